// RNAFold_31430570672224
// MI455X (gfx1250) — compile-verified
//
#include <hip/hip_runtime.h>

typedef __attribute__((ext_vector_type(2))) float v2f;
typedef __attribute__((ext_vector_type(8))) float v8f;

#define BATCH   2048
#define TILES   (BATCH / 16)   // 128 tiles of 16 batch rows
#define CHUNK_K 256            // K elements per wave-chunk (64 WMMAs)

// ---------------------------------------------------------------------------
// s[b] = 0
// ---------------------------------------------------------------------------
__global__ void zero_s_kernel(float* __restrict__ s) {
    int i = blockIdx.x * blockDim.x + threadIdx.x;
    if (i < BATCH) s[i] = 0.0f;
}

// ---------------------------------------------------------------------------
// One wave (32 threads) per block. Accumulate
//   s[row] += dot(C[row, k0:klim], W[k0:klim])   for a 16-row batch tile,
// using V_WMMA_F32_16X16X4_F32.
//
// A layout (16x4 f32, ISA 7.12.2): lane L<16 holds A[M=L][K=0],A[M=L][K=1];
// lane 16+L holds A[M=L][K=2],A[M=L][K=3]. B (4x16) mirrors this in K.
//
// Key trick: B carries the weights broadcast to EVERY column, B[k][n]=w[k].
// Then D[m][n] = dot(row m) for every n — all 16 columns identical — so the
// B operand needs no lane masking at all: b = {hib? w2:w0, hib? w3:w1}.
// All four (wave-uniform, SMEM-loaded) weights are used unconditionally in
// all lanes, so the compiler cannot predicate/sink the loads, and the loop
// body stays branch-free with EXEC all-1s (required by WMMA).
//
// D readout: lane 0 holds D[M=0..7][0], lane 16 holds D[M=8..15][0].
// ---------------------------------------------------------------------------
__global__ __launch_bounds__(32)
void wmma_dot_kernel(const float* __restrict__ c, const float* __restrict__ w,
                     int K, float* __restrict__ s) {
    const int  lane  = threadIdx.x;      // 0..31
    const int  m     = lane & 15;        // row within tile / N index
    const int  hi    = lane >> 4;        // 0: K pair {0,1}, 1: K pair {2,3}
    const int  khalf = hi * 2;
    const bool hib   = (hi != 0);

    const int tile = blockIdx.x;                 // 0..TILES-1
    const int k0   = blockIdx.y * CHUNK_K;       // scalar chunk start
    if (k0 >= K) return;
    const int klim  = (k0 + CHUNK_K < K) ? (k0 + CHUNK_K) : K;
    const int kmain = k0 + ((klim - k0) & ~3);   // full 4-wide quads

    const int row = tile * 16 + m;
    const float* __restrict__ crow = c + (size_t)row * (size_t)K + khalf;

    v8f acc = {};

    int k = k0;
    #pragma unroll 4
    for (; k < kmain; k += 4) {
        // A: two consecutive counts from this lane's batch row (vector b64).
        v2f a;
        a.x = crow[k];
        a.y = crow[k + 1];
        // Uniform weight quad (scalar/SMEM path, 16B aligned: k % 4 == 0).
        const float w0 = w[k];
        const float w1 = w[k + 1];
        const float w2 = w[k + 2];
        const float w3 = w[k + 3];
        // Broadcast weights to all B columns: select by K-half only.
        v2f b;
        b.x = hib ? w2 : w0;
        b.y = hib ? w3 : w1;
        acc = __builtin_amdgcn_wmma_f32_16x16x4_f32(
                  /*neg_a=*/false, a, /*neg_b=*/false, b,
                  /*c_mod=*/(short)0, acc, /*reuse_a=*/false, /*reuse_b=*/false);
    }
    if (k < klim) {                      // guarded tail quad (table end only)
        const int ka = k + khalf;        // absolute K index for this lane
        v2f a;
        a.x = (ka     < K) ? crow[k]     : 0.0f;
        a.y = (ka + 1 < K) ? crow[k + 1] : 0.0f;
        const float w0 = w[k];
        const float w1 = (k + 1 < K) ? w[k + 1] : 0.0f;
        const float w2 = (k + 2 < K) ? w[k + 2] : 0.0f;
        const float w3 = (k + 3 < K) ? w[k + 3] : 0.0f;
        v2f b;
        b.x = hib ? w2 : w0;
        b.y = hib ? w3 : w1;
        acc = __builtin_amdgcn_wmma_f32_16x16x4_f32(
                  false, a, false, b, (short)0, acc, false, false);
    }

    // Column 0 of D: lane 0 -> rows 0..7, lane 16 -> rows 8..15.
    if (m == 0) {
        float* __restrict__ sb = s + tile * 16 + hi * 8;
        #pragma unroll
        for (int j = 0; j < 8; ++j)
            atomicAdd(&sb[j], acc[j]);
    }
}

// ---------------------------------------------------------------------------
// Fold in the 8 scalar tables and apply out = s + (v - s)  (value == v,
// matching the reference's straight-through estimator arithmetic).
// ---------------------------------------------------------------------------
__global__ void finalize_kernel(const float* __restrict__ s, const float* __restrict__ v,
                                const float* __restrict__ c0, const float* __restrict__ w0,
                                const float* __restrict__ c1, const float* __restrict__ w1,
                                const float* __restrict__ c2, const float* __restrict__ w2,
                                const float* __restrict__ c3, const float* __restrict__ w3,
                                const float* __restrict__ c4, const float* __restrict__ w4,
                                const float* __restrict__ c5, const float* __restrict__ w5,
                                const float* __restrict__ c6, const float* __restrict__ w6,
                                const float* __restrict__ c7, const float* __restrict__ w7,
                                float* __restrict__ out) {
    int b = blockIdx.x * blockDim.x + threadIdx.x;
    if (b >= BATCH) return;
    float acc = s[b];
    acc = fmaf(c0[b], w0[0], acc);
    acc = fmaf(c1[b], w1[0], acc);
    acc = fmaf(c2[b], w2[0], acc);
    acc = fmaf(c3[b], w3[0], acc);
    acc = fmaf(c4[b], w4[0], acc);
    acc = fmaf(c5[b], w5[0], acc);
    acc = fmaf(c6[b], w6[0], acc);
    acc = fmaf(c7[b], w7[0], acc);
    float t = v[b] - acc;               // out = s + (v - s), fp32 order preserved
    out[b] = acc + t;
}

// ---------------------------------------------------------------------------
// Host side. d_in dict order: (c_tbl, w_tbl) pairs for the 23 tables, then v.
// ---------------------------------------------------------------------------
extern "C" void kernel_launch(void* const* d_in, const int* in_sizes, int n_in,
                              void* d_out, int out_size, void* d_ws, size_t ws_size,
                              hipStream_t stream) {
    (void)in_sizes; (void)n_in; (void)out_size; (void)ws_size;

    // Per-table K (flattened table element count), _TABLES order.
    static const int tblK[23] = {
        64,            // stack      8*8
        31, 31, 31,    // hairpin, bulge, internal
        200, 200, 200, 200, 200, 200,  // mm_* 8*5*5
        1600,          // int11      8*8*5*5
        8000,          // int21      8*8*5*5*5
        30625,         // int22      7*7*5*5*5*5
        40, 40,        // dangle5, dangle3
        1, 1, 1, 1, 1, 1, 1, 1         // 8 scalar tables
    };

    float* s = (float*)d_ws;                      // [BATCH] partial sums
    const float* v = (const float*)d_in[46];      // last dict entry
    float* out = (float*)d_out;

    zero_s_kernel<<<(BATCH + 255) / 256, 256, 0, stream>>>(s);

    // Tables 0..14 (K > 1) go through the WMMA GEMV path.
    for (int t = 0; t < 15; ++t) {
        const int K = tblK[t];
        const int chunks = (K + CHUNK_K - 1) / CHUNK_K;
        dim3 grid(TILES, chunks);
        wmma_dot_kernel<<<grid, 32, 0, stream>>>(
            (const float*)d_in[2 * t], (const float*)d_in[2 * t + 1], K, s);
    }

    // Tables 15..22 are scalars; fold into the finalize pass.
    finalize_kernel<<<(BATCH + 255) / 256, 256, 0, stream>>>(
        s, v,
        (const float*)d_in[30], (const float*)d_in[31],
        (const float*)d_in[32], (const float*)d_in[33],
        (const float*)d_in[34], (const float*)d_in[35],
        (const float*)d_in[36], (const float*)d_in[37],
        (const float*)d_in[38], (const float*)d_in[39],
        (const float*)d_in[40], (const float*)d_in[41],
        (const float*)d_in[42], (const float*)d_in[43],
        (const float*)d_in[44], (const float*)d_in[45],
        out);
}